// FeatureGenerator_60078002536843
// MI455X (gfx1250) — compile-verified
//
#include <hip/hip_runtime.h>
#include <math.h>

#ifndef __has_builtin
#define __has_builtin(x) 0
#endif

// gfx1250 async LDS DMA path (probe via __has_builtin; fallback = plain copies)
#if __has_builtin(__builtin_amdgcn_global_load_async_to_lds_b128) && \
    __has_builtin(__builtin_amdgcn_global_store_async_from_lds_b128)
#define USE_ASYNC 1
#else
#define USE_ASYNC 0
#endif

#define AS1 __attribute__((address_space(1)))
#define AS3 __attribute__((address_space(3)))

typedef int v4i __attribute__((vector_size(16)));   // matches builtin param type

namespace {
constexpr int   Tt   = 2048;  // time steps
constexpr int   NF   = 28;    // floats per (b,t) input row (M*K*C)
constexpr int   NP   = 14;    // points (M*K)
constexpr int   F    = 181;   // output features: 28 + 91 + 28 + 28 + 6
constexpr int   TBLK = 64;    // t rows per block
constexpr float EPSF = 1e-6f;
}

__global__ __launch_bounds__(TBLK)
void feat_kernel(const float* __restrict__ x, float* __restrict__ out) {
  // 66 rows of 28 floats (112B rows -> 16B aligned) + 64 rows of 181 output floats
  __shared__ __align__(16) float xs[(TBLK + 2) * NF];     //  7,392 B
  __shared__ __align__(16) float outs[TBLK * F];          // 46,336 B

  const int tid = threadIdx.x;
  const int b   = blockIdx.x >> 5;            // Tt/TBLK == 32 chunks per b
  const int t0  = (blockIdx.x & 31) * TBLK;
  const int t   = t0 + tid;

  // ---- stage x[b, t0-2 .. t0+TBLK-1, :] into LDS (async DMA on CDNA5) ----
  for (int q = tid; q < (TBLK + 2) * (NF / 4); q += TBLK) {
    const int r  = q / (NF / 4);
    const int c  = (q - r * (NF / 4)) * 4;
    const int tr = t0 - 2 + r;
    if (tr >= 0) {                            // skip halo rows before t=0
      const float* src = x + ((size_t)b * Tt + tr) * NF + c;
      float*       dst = &xs[r * NF + c];
#if USE_ASYNC
      __builtin_amdgcn_global_load_async_to_lds_b128((AS1 v4i*)src,
                                                     (AS3 v4i*)dst, 0, 0);
#else
      *(float4*)dst = *(const float4*)src;
#endif
    }
  }
#if USE_ASYNC
#if __has_builtin(__builtin_amdgcn_s_wait_asynccnt)
  __builtin_amdgcn_s_wait_asynccnt(0);
#else
  asm volatile("s_wait_asynccnt 0" ::: "memory");
#endif
#endif
  __syncthreads();

  // ---- per-thread feature computation (one (b,t) per thread) ----
  float xc[NF], xm1[NF], xm2[NF];
#pragma unroll
  for (int k = 0; k < NF; ++k) {
    xc[k]  = xs[(tid + 2) * NF + k];
    xm1[k] = xs[(tid + 1) * NF + k];
    xm2[k] = xs[(tid + 0) * NF + k];
  }

  float* od = &outs[tid * F];   // stride 181 words (odd) -> conflict-free

  // [0,28): raw x
#pragma unroll
  for (int k = 0; k < NF; ++k) od[k] = xc[k];

  // [28,119): pairwise distances, np.triu_indices(14, k=1) row-major order
  {
    int n = NF;
#pragma unroll
    for (int i = 0; i < NP; ++i)
#pragma unroll
      for (int j = i + 1; j < NP; ++j) {
        const float dx = xc[2 * i]     - xc[2 * j];
        const float dy = xc[2 * i + 1] - xc[2 * j + 1];
        od[n++] = sqrtf(dx * dx + dy * dy);
      }
  }

  // [119,147): v ; [147,175): a   (v[0]=0, a[0]=0, a[1]=v[1])
  {
    const bool h1 = (t >= 1), h2 = (t >= 2);
#pragma unroll
    for (int k = 0; k < NF; ++k) {
      const float vk = h1 ? (xc[k] - xm1[k])  : 0.0f;
      const float vm = h2 ? (xm1[k] - xm2[k]) : 0.0f;
      od[119 + k] = vk;
      od[147 + k] = h1 ? (vk - vm) : 0.0f;
    }
  }

  // [175,181): body = [bl0, blc0, curv0, bl1, blc1, curv1]
  {
    float bl[2], cv[2];
#pragma unroll
    for (int m = 0; m < 2; ++m) {
      const int o = m * 14;                  // (m, k=0, c=0) flat offset
      const float nx = xc[o + 0],  ny = xc[o + 1];    // nose  (k=0)
      const float cx = xc[o + 6],  cy = xc[o + 7];    // neck  (k=3)
      const float tx = xc[o + 12], ty = xc[o + 13];   // tail  (k=6)
      const float dbx = nx - tx, dby = ny - ty;
      bl[m] = sqrtf(dbx * dbx + dby * dby);
      const float v1x = nx - cx, v1y = ny - cy;
      const float v2x = tx - cx, v2y = ty - cy;
      const float r1 = 1.0f / (sqrtf(v1x * v1x + v1y * v1y) + EPSF);
      const float r2 = 1.0f / (sqrtf(v2x * v2x + v2y * v2y) + EPSF);
      float c = (v1x * v2x + v1y * v2y) * r1 * r2;
      c = fminf(fmaxf(c, -1.0f + EPSF), 1.0f - EPSF);
      cv[m] = acosf(c);
    }
    od[175] = bl[0]; od[176] = 0.0f;          od[177] = cv[0];
    od[178] = bl[1]; od[179] = bl[1] - bl[0]; od[180] = cv[1];
  }

  __syncthreads();

  // ---- coalesced writeback of the contiguous 64x181 tile (async from LDS) ----
  float* gbase = out + ((size_t)b * Tt + t0) * F;
  constexpr int NQ = TBLK * F / 4;            // 2896 float4s
  for (int q = tid; q < NQ; q += TBLK) {
#if USE_ASYNC
    __builtin_amdgcn_global_store_async_from_lds_b128(
        (AS1 v4i*)(gbase + q * 4), (AS3 v4i*)&outs[q * 4], 0, 0);
#else
    *(float4*)(gbase + q * 4) = *(const float4*)&outs[q * 4];
#endif
  }
  // s_endpgm performs an implicit wait-idle, draining ASYNCcnt.
}

extern "C" void kernel_launch(void* const* d_in, const int* in_sizes, int n_in,
                              void* d_out, int out_size, void* d_ws, size_t ws_size,
                              hipStream_t stream) {
  const float* x  = (const float*)d_in[0];
  float* out      = (float*)d_out;
  const int B     = in_sizes[0] / (Tt * NF);  // 64 for the reference shapes
  dim3 grid(B * (Tt / TBLK));                 // 2048 blocks
  dim3 block(TBLK);                           // 64 threads = 2 waves
  feat_kernel<<<grid, block, 0, stream>>>(x, out);
}